// SGC_43533788512788
// MI455X (gfx1250) — compile-verified
//
#include <hip/hip_runtime.h>
#include <hip/hip_bf16.h>

typedef __attribute__((ext_vector_type(2))) float v2f;
typedef __attribute__((ext_vector_type(8))) float v8f;

#define HID_N 128

// ---------------- degree / normalization ----------------

__global__ void deg_init_kernel(float* __restrict__ deg, int n) {
    int i = blockIdx.x * blockDim.x + threadIdx.x;
    if (i < n) deg[i] = 1.0f;  // self loop
}

__global__ void deg_count_kernel(const int* __restrict__ col, float* __restrict__ deg, int e) {
    int i = blockIdx.x * blockDim.x + threadIdx.x;
    if (i < e) atomicAdd(&deg[col[i]], 1.0f);
}

__global__ void dinv_kernel(float* __restrict__ deg, int n) {
    int i = blockIdx.x * blockDim.x + threadIdx.x;
    if (i < n) {
        float d = deg[i];
        deg[i] = (d > 0.0f) ? rsqrtf(d) : 0.0f;  // deg >= 1 always, but match ref
    }
}

// ---------------- dense layer: Y[M,128] = act(X[M,K] @ W[K,128] + b) ----------------
// One wave computes a 16-row x 128-col strip using V_WMMA_F32_16X16X4_F32.
// A 16x4 f32 layout: lane l (l<16): {A[l][kb], A[l][kb+1]}; lanes 16-31: {A[l-16][kb+2], A[l-16][kb+3]}
// B 4x16 f32 layout: v0 = rows kb / kb+2 striped over lanes, v1 = rows kb+1 / kb+3
// C/D 16x16 f32:     VGPR v = row v (lanes 0-15) / row v+8 (lanes 16-31), col = lane&15
__global__ __launch_bounds__(256) void gemm_wmma_kernel(
    const float* __restrict__ X, const float* __restrict__ W,
    const float* __restrict__ bias, float* __restrict__ Y,
    int M, int K, int leaky) {
    int wid  = (blockIdx.x * blockDim.x + threadIdx.x) >> 5;
    int lane = threadIdx.x & 31;
    int m0 = wid * 16;
    if (m0 >= M) return;  // wave-uniform: EXEC stays all-1s for WMMA

    int lane_lo = lane & 15;
    int lane_hi = lane >> 4;

    v8f acc[8];
#pragma unroll
    for (int t = 0; t < 8; ++t) acc[t] = (v8f)(0.0f);

    const float* xrow = X + (size_t)(m0 + lane_lo) * (size_t)K;
    int ksteps = K >> 2;
    for (int ks = 0; ks < ksteps; ++ks) {
        int kb = (ks << 2) + lane_hi * 2;
        v2f a;
        a.x = xrow[kb];
        a.y = xrow[kb + 1];
        const float* wrow = W + (size_t)kb * HID_N;
#pragma unroll
        for (int nt = 0; nt < 8; ++nt) {
            v2f b;
            b.x = wrow[nt * 16 + lane_lo];
            b.y = wrow[HID_N + nt * 16 + lane_lo];
            acc[nt] = __builtin_amdgcn_wmma_f32_16x16x4_f32(
                /*neg_a=*/false, a, /*neg_b=*/false, b,
                /*c_mod=*/(short)0, acc[nt], /*reuse_a=*/false, /*reuse_b=*/false);
        }
    }

#pragma unroll
    for (int nt = 0; nt < 8; ++nt) {
        float bv = bias[nt * 16 + lane_lo];
#pragma unroll
        for (int v = 0; v < 8; ++v) {
            int m = m0 + v + lane_hi * 8;
            float val = acc[nt][v] + bv;
            if (leaky) val = (val >= 0.0f) ? val : 0.01f * val;
            Y[(size_t)m * HID_N + nt * 16 + lane_lo] = val;
        }
    }
}

// ---------------- propagate: agg = scatter_add(norm * x[row] -> col) ----------------

// self-loop contribution doubles as the zero-init of agg
__global__ void prop_init_kernel(const float* __restrict__ x, const float* __restrict__ dinv,
                                 float* __restrict__ agg, int n) {
    int i = blockIdx.x * blockDim.x + threadIdx.x;
    int total = n * HID_N;
    if (i < total) {
        int node = i >> 7;  // /128
        float di = dinv[node];
        agg[i] = di * di * x[i];
    }
}

// one wave per edge; lane handles 4 consecutive features via float4
__global__ void prop_edges_kernel(const int* __restrict__ row, const int* __restrict__ col,
                                  const float* __restrict__ dinv, const float* __restrict__ x,
                                  float* __restrict__ agg, int e) {
    int wid  = (blockIdx.x * blockDim.x + threadIdx.x) >> 5;
    int lane = threadIdx.x & 31;
    if (wid >= e) return;
    int s = row[wid];
    int d = col[wid];
    float w = dinv[s] * dinv[d];
    const float4* xs = (const float4*)(x + (size_t)s * HID_N);
    float4 v = xs[lane];
    float* out = agg + (size_t)d * HID_N + lane * 4;
    atomicAdd(out + 0, w * v.x);
    atomicAdd(out + 1, w * v.y);
    atomicAdd(out + 2, w * v.z);
    atomicAdd(out + 3, w * v.w);
}

// ---------------- output layer: y[M,3] = x[M,128] @ Wout[128,3] + bout ----------------

__global__ void out_kernel(const float* __restrict__ x, const float* __restrict__ Wout,
                           const float* __restrict__ bout, float* __restrict__ y, int n) {
    int i = blockIdx.x * blockDim.x + threadIdx.x;
    if (i >= n) return;
    float a0 = bout[0], a1 = bout[1], a2 = bout[2];
    const float* xr = x + (size_t)i * HID_N;
#pragma unroll 4
    for (int h = 0; h < HID_N; ++h) {
        float v = xr[h];
        a0 += v * Wout[h * 3 + 0];
        a1 += v * Wout[h * 3 + 1];
        a2 += v * Wout[h * 3 + 2];
    }
    y[(size_t)i * 3 + 0] = a0;
    y[(size_t)i * 3 + 1] = a1;
    y[(size_t)i * 3 + 2] = a2;
}

// ---------------- driver ----------------

extern "C" void kernel_launch(void* const* d_in, const int* in_sizes, int n_in,
                              void* d_out, int out_size, void* d_ws, size_t ws_size,
                              hipStream_t stream) {
    const float* feature = (const float*)d_in[0];   // [N,16]
    const int*   eidx    = (const int*)d_in[1];     // [2,E]
    // d_in[2] edge_type: unused in forward
    const float* W_in  = (const float*)d_in[3];     // [16,128]
    const float* b_in  = (const float*)d_in[4];     // [128]
    const float* W1    = (const float*)d_in[5];     // [128,128]
    const float* b1    = (const float*)d_in[6];     // [128]
    const float* W2    = (const float*)d_in[7];     // [128,128]
    const float* b2    = (const float*)d_in[8];     // [128]
    const float* W_out = (const float*)d_in[9];     // [128,3]
    const float* b_out = (const float*)d_in[10];    // [3]
    float* out = (float*)d_out;

    const int N = in_sizes[0] / 16;
    const int E = in_sizes[2];
    const int* erow = eidx;
    const int* ecol = eidx + E;

    // workspace: dinv | bufA | bufB
    size_t off_dinv = 0;
    size_t off_A = ((size_t)N * 4 + 255) / 256 * 256;
    size_t off_B = off_A + (size_t)N * HID_N * 4;
    float* dinv = (float*)((char*)d_ws + off_dinv);
    float* bufA = (float*)((char*)d_ws + off_A);
    float* bufB = (float*)((char*)d_ws + off_B);

    const int TB = 256;
    int gN   = (N + TB - 1) / TB;
    int gE   = (E + TB - 1) / TB;
    int gNH  = (N * HID_N + TB - 1) / TB;
    int nWavesM = (N + 15) / 16;                       // one wave per 16 rows
    int gGemm = (nWavesM * 32 + TB - 1) / TB;
    int gEdge = ((size_t)E * 32 + TB - 1) / TB;        // one wave per edge

    // 1) GCN normalization: deg = 1 + in-degree(col); dinv = rsqrt(deg)
    deg_init_kernel<<<gN, TB, 0, stream>>>(dinv, N);
    deg_count_kernel<<<gE, TB, 0, stream>>>(ecol, dinv, E);
    dinv_kernel<<<gN, TB, 0, stream>>>(dinv, N);

    // 2) x0 = leaky_relu(feature @ W_in + b_in)
    gemm_wmma_kernel<<<gGemm, TB, 0, stream>>>(feature, W_in, b_in, bufA, N, 16, 1);

    // 3) agg1 = propagate(x0); x1 = agg1 @ W1 + b1
    prop_init_kernel<<<gNH, TB, 0, stream>>>(bufA, dinv, bufB, N);
    prop_edges_kernel<<<gEdge, TB, 0, stream>>>(erow, ecol, dinv, bufA, bufB, E);
    gemm_wmma_kernel<<<gGemm, TB, 0, stream>>>(bufB, W1, b1, bufA, N, HID_N, 0);

    // 4) agg2 = propagate(x1); x2 = agg2 @ W2 + b2
    prop_init_kernel<<<gNH, TB, 0, stream>>>(bufA, dinv, bufB, N);
    prop_edges_kernel<<<gEdge, TB, 0, stream>>>(erow, ecol, dinv, bufA, bufB, E);
    gemm_wmma_kernel<<<gGemm, TB, 0, stream>>>(bufB, W2, b2, bufA, N, HID_N, 0);

    // 5) out = x2 @ W_out + b_out
    out_kernel<<<gN, TB, 0, stream>>>(bufA, W_out, b_out, out, N);
}